// ResGCN_31353261261180
// MI455X (gfx1250) — compile-verified
//
#include <hip/hip_runtime.h>

#define N_NODES  50000
#define N_EDGES  1250000
#define D_HID    64
#define D_OUT    16
#define N_GRAPHS 128

typedef __attribute__((ext_vector_type(2))) float v2f;
typedef __attribute__((ext_vector_type(8))) float v8f;

// ---------------------------------------------------------------------------
// Zero-fill
// ---------------------------------------------------------------------------
__global__ void fill_zero(float* __restrict__ p, int n) {
    int i = blockIdx.x * blockDim.x + threadIdx.x;
    if (i < n) p[i] = 0.0f;
}

// ---------------------------------------------------------------------------
// Edge scatter-add: agg[dst] += h[src]  (64 f32 per edge, float4 per thread)
// Working set (h, agg) = 2 x 12.8 MB -> resident in 192 MB L2.
// ---------------------------------------------------------------------------
__global__ void scatter_edges(const float* __restrict__ h,
                              const int* __restrict__ src,
                              const int* __restrict__ dst,
                              float* __restrict__ agg) {
    int idx = blockIdx.x * blockDim.x + threadIdx.x;   // N_EDGES * 16 threads
    if (idx >= N_EDGES * 16) return;
    int e = idx >> 4;
    int q = idx & 15;
    int s = src[e];
    int d = dst[e];
    float4 v = ((const float4*)(h + (size_t)s * 64))[q];
    float* o = agg + (size_t)d * 64 + q * 4;
    unsafeAtomicAdd(o + 0, v.x);
    unsafeAtomicAdd(o + 1, v.y);
    unsafeAtomicAdd(o + 2, v.z);
    unsafeAtomicAdd(o + 3, v.w);
}

// ---------------------------------------------------------------------------
// Fused GraphConv GEMM via V_WMMA_F32_16X16X4_F32:
//   out = maybe_relu( agg @ Wrel + x @ Wroot + bias )
// Block = 128 threads = 4 waves; wave w handles col block w of a 16-row stripe.
// K=64 per operand -> 16 chained k=4 WMMAs per operand, 32 total.
// ---------------------------------------------------------------------------
__global__ __launch_bounds__(128) void gemm_fused(
        const float* __restrict__ agg, const float* __restrict__ xin,
        const float* __restrict__ Wrel, const float* __restrict__ Wroot,
        const float* __restrict__ bias, float* __restrict__ out, int relu)
{
    const int lane   = threadIdx.x & 31;
    const int colBlk = threadIdx.x >> 5;          // 0..3
    const int rowBlk = blockIdx.x;                // 0..3124 (50000/16)
    const int m   = lane & 15;
    const int row = rowBlk * 16 + m;              // A-matrix row for this lane
    const int col = colBlk * 16 + m;              // B/C/D column for this lane
    const int kh  = (lane >> 4) << 1;             // 0 for lanes 0-15, 2 for 16-31

    const float* arow = agg + (size_t)row * 64 + kh;
    const float* xrow = xin + (size_t)row * 64 + kh;

    v8f c = {};

    #pragma unroll
    for (int kt = 0; kt < 16; ++kt) {             // agg @ Wrel, K = 64
        v2f a = *(const v2f*)(arow + kt * 4);     // A[row][k0+kh], A[row][k0+kh+1]
        v2f b;
        b.x = Wrel[(kt * 4 + kh)     * 64 + col]; // B[k0+kh][col]
        b.y = Wrel[(kt * 4 + kh + 1) * 64 + col]; // B[k0+kh+1][col]
        c = __builtin_amdgcn_wmma_f32_16x16x4_f32(false, a, false, b,
                                                  (short)0, c, false, false);
    }
    #pragma unroll
    for (int kt = 0; kt < 16; ++kt) {             // x @ Wroot, K = 64
        v2f a = *(const v2f*)(xrow + kt * 4);
        v2f b;
        b.x = Wroot[(kt * 4 + kh)     * 64 + col];
        b.y = Wroot[(kt * 4 + kh + 1) * 64 + col];
        c = __builtin_amdgcn_wmma_f32_16x16x4_f32(false, a, false, b,
                                                  (short)0, c, false, false);
    }

    const float bc = bias[col];
    const int rbase = rowBlk * 16 + ((lane >> 4) << 3);  // rows r / r+8 split
    #pragma unroll
    for (int r = 0; r < 8; ++r) {
        float v = c[r] + bc;
        if (relu) v = fmaxf(v, 0.0f);
        out[(size_t)(rbase + r) * 64 + col] = v;
    }
}

// ---------------------------------------------------------------------------
// Global mean-pool: sums[batch[n]] += h[n]; counts[batch[n]] += 1
// ---------------------------------------------------------------------------
__global__ void pool_scatter(const float* __restrict__ h,
                             const int* __restrict__ batch,
                             float* __restrict__ sums,
                             float* __restrict__ cnts) {
    int idx = blockIdx.x * blockDim.x + threadIdx.x;   // N_NODES * 16 threads
    if (idx >= N_NODES * 16) return;
    int node = idx >> 4;
    int q    = idx & 15;
    int g    = batch[node];
    float4 v = ((const float4*)(h + (size_t)node * 64))[q];
    float* s = sums + (size_t)g * 64 + q * 4;
    unsafeAtomicAdd(s + 0, v.x);
    unsafeAtomicAdd(s + 1, v.y);
    unsafeAtomicAdd(s + 2, v.z);
    unsafeAtomicAdd(s + 3, v.w);
    if (q == 0) unsafeAtomicAdd(cnts + g, 1.0f);
}

// ---------------------------------------------------------------------------
// pooled @ W_lin + b_lin  (128x16 output, K=64) — trivially small
// ---------------------------------------------------------------------------
__global__ void final_linear(const float* __restrict__ sums,
                             const float* __restrict__ cnts,
                             const float* __restrict__ Wlin,
                             const float* __restrict__ blin,
                             float* __restrict__ out) {
    int idx = blockIdx.x * blockDim.x + threadIdx.x;
    if (idx >= N_GRAPHS * D_OUT) return;
    int g = idx / D_OUT;
    int o = idx - g * D_OUT;
    float inv = 1.0f / fmaxf(cnts[g], 1.0f);
    float acc = blin[o];
    #pragma unroll 8
    for (int k = 0; k < 64; ++k)
        acc += (sums[(size_t)g * 64 + k] * inv) * Wlin[k * D_OUT + o];
    out[idx] = acc;
}

// ---------------------------------------------------------------------------
extern "C" void kernel_launch(void* const* d_in, const int* in_sizes, int n_in,
                              void* d_out, int out_size, void* d_ws, size_t ws_size,
                              hipStream_t stream) {
    const float* x      = (const float*)d_in[0];
    const int*   ei     = (const int*)  d_in[1];
    const int*   batch  = (const int*)  d_in[3];
    const float* Wrel1  = (const float*)d_in[4];
    const float* brel1  = (const float*)d_in[5];
    const float* Wroot1 = (const float*)d_in[6];
    const float* Wrel2  = (const float*)d_in[7];
    const float* brel2  = (const float*)d_in[8];
    const float* Wroot2 = (const float*)d_in[9];
    const float* Wrel3  = (const float*)d_in[10];
    const float* brel3  = (const float*)d_in[11];
    const float* Wroot3 = (const float*)d_in[12];
    const float* Wlin   = (const float*)d_in[13];
    const float* blin   = (const float*)d_in[14];
    float* out = (float*)d_out;

    const size_t NB = (size_t)N_NODES * 64 * sizeof(float);   // 12.8 MB
    char* ws = (char*)d_ws;
    float* agg  = (float*)(ws);
    float* hA   = (float*)(ws + NB);
    float* hB   = (float*)(ws + 2 * NB);
    float* sums = (float*)(ws + 3 * NB);
    float* cnts = sums + N_GRAPHS * 64;

    const int* src = ei;
    const int* dst = ei + N_EDGES;

    const int NF = N_NODES * 64;                       // 3.2M floats
    dim3 b256(256);
    dim3 fillG((NF + 255) / 256);
    dim3 scatG((N_EDGES * 16 + 255) / 256);            // 20M threads
    dim3 gemmG(N_NODES / 16);                          // 3125 blocks
    dim3 gemmB(128);
    dim3 poolG((N_NODES * 16 + 255) / 256);

    // Layer 1: h1 = relu(agg(x) @ Wrel1 + b1 + x @ Wroot1)
    fill_zero    <<<fillG, b256, 0, stream>>>(agg, NF);
    scatter_edges<<<scatG, b256, 0, stream>>>(x, src, dst, agg);
    gemm_fused   <<<gemmG, gemmB, 0, stream>>>(agg, x, Wrel1, Wroot1, brel1, hA, 1);

    // Layer 2
    fill_zero    <<<fillG, b256, 0, stream>>>(agg, NF);
    scatter_edges<<<scatG, b256, 0, stream>>>(hA, src, dst, agg);
    gemm_fused   <<<gemmG, gemmB, 0, stream>>>(agg, hA, Wrel2, Wroot2, brel2, hB, 1);

    // Layer 3 (no relu)
    fill_zero    <<<fillG, b256, 0, stream>>>(agg, NF);
    scatter_edges<<<scatG, b256, 0, stream>>>(hB, src, dst, agg);
    gemm_fused   <<<gemmG, gemmB, 0, stream>>>(agg, hB, Wrel3, Wroot3, brel3, hA, 0);

    // Mean pool + final linear
    fill_zero   <<<dim3((N_GRAPHS * 64 + N_GRAPHS + 255) / 256), b256, 0, stream>>>(
                    sums, N_GRAPHS * 64 + N_GRAPHS);
    pool_scatter<<<poolG, b256, 0, stream>>>(hA, batch, sums, cnts);
    final_linear<<<dim3((N_GRAPHS * D_OUT + 255) / 256), b256, 0, stream>>>(
                    sums, cnts, Wlin, blin, out);
}